// ChannleWiseMinMaxStats_13855564497198
// MI455X (gfx1250) — compile-verified
//
#include <hip/hip_runtime.h>
#include <hip/hip_bf16.h>
#include <float.h>

#define THREADS 256
#define STAGES 8                       // 8 x 4KB block-steps in flight
#define STEP_BYTES (THREADS * 16)      // 4096 B per block-step

// Async global->LDS 128-bit copy (CDNA5 GLOBAL_LOAD_ASYNC_TO_LDS_B128, GVS mode:
// 64-bit SGPR base + per-lane 32-bit byte offset -> per-lane LDS byte offset).
// Tracked by ASYNCcnt.
__device__ __forceinline__ void async_copy_b128(unsigned lds_byte_off,
                                                unsigned g_byte_off,
                                                const float* sbase) {
    asm volatile("global_load_async_to_lds_b128 %0, %1, %2"
                 :
                 : "v"(lds_byte_off), "v"(g_byte_off), "s"(sbase)
                 : "memory");
}

__device__ __forceinline__ void acc_minmax(float4 v, float& mn, float& mx) {
    mn = fminf(mn, fminf(fminf(v.x, v.y), fminf(v.z, v.w)));
    mx = fmaxf(mx, fmaxf(fmaxf(v.x, v.y), fmaxf(v.z, v.w)));
}

__global__ __launch_bounds__(THREADS)
void ChannelWiseMinMax_kernel(const float* __restrict__ in,
                              float* __restrict__ out,
                              int C, int N) {
    __shared__ float4 buf[STAGES][THREADS];     // 32 KB staging
    __shared__ float  wmin[THREADS / 32];
    __shared__ float  wmax[THREADS / 32];

    const int c   = blockIdx.x;
    const int tid = threadIdx.x;
    const float* row = in + (size_t)c * (size_t)N;

    const int steps = N >> 10;                  // 1024 floats per block-step
    const int full  = steps & ~(STAGES - 1);    // steps covered by the pipeline

    float mn = FLT_MAX, mx = -FLT_MAX;

    if (full >= STAGES) {
        const unsigned goff0 = (unsigned)(tid * 16);

        // ---- prologue: fill all 8 stages (compile-time stage indices) ----
#pragma unroll
        for (int k = 0; k < STAGES; ++k)
            async_copy_b128((unsigned)(uintptr_t)(&buf[k][tid]),
                            goff0 + (unsigned)k * STEP_BYTES, row);

        // ---- steady state: wait oldest -> consume -> refill same stage ----
        const int main_iters = full - STAGES;
        for (int i = 0; i < main_iters; i += STAGES) {
            const unsigned next = goff0 + (unsigned)(i + STAGES) * STEP_BYTES;
#pragma unroll
            for (int k = 0; k < STAGES; ++k) {
                asm volatile("s_wait_asynccnt 0x7" ::: "memory");
                float4 v = buf[k][tid];
                acc_minmax(v, mn, mx);
                async_copy_b128((unsigned)(uintptr_t)(&buf[k][tid]),
                                next + (unsigned)k * STEP_BYTES, row);
            }
        }

        // ---- drain: everything outstanding, then consume last 8 stages ----
        asm volatile("s_wait_asynccnt 0x0" ::: "memory");
#pragma unroll
        for (int k = 0; k < STAGES; ++k) {
            float4 v = buf[k][tid];
            acc_minmax(v, mn, mx);
        }
    }

    // ---- remainder (N not a multiple of STAGES*1024): plain coalesced ----
    for (int j = full * (THREADS * 4) + tid; j < N; j += THREADS) {
        float v = row[j];
        mn = fminf(mn, v);
        mx = fmaxf(mx, v);
    }

    // ---- wave32 butterfly reduction (gfx1250 warpSize == 32) ----
#pragma unroll
    for (int off = 16; off > 0; off >>= 1) {
        mn = fminf(mn, __shfl_xor(mn, off, 32));
        mx = fmaxf(mx, __shfl_xor(mx, off, 32));
    }
    if ((tid & 31) == 0) {
        wmin[tid >> 5] = mn;
        wmax[tid >> 5] = mx;
    }
    __syncthreads();

    if (tid == 0) {
        float bm = wmin[0], bM = wmax[0];
#pragma unroll
        for (int w = 1; w < THREADS / 32; ++w) {
            bm = fminf(bm, wmin[w]);
            bM = fmaxf(bM, wmax[w]);
        }
        out[c]     = bm;   // min_vals
        out[C + c] = bM;   // max_vals
    }
}

extern "C" void kernel_launch(void* const* d_in, const int* in_sizes, int n_in,
                              void* d_out, int out_size, void* d_ws, size_t ws_size,
                              hipStream_t stream) {
    (void)n_in; (void)d_ws; (void)ws_size;
    const float* in = (const float*)d_in[0];
    float* out = (float*)d_out;

    const int C = out_size / 2;             // 1024 channels
    const int N = in_sizes[0] / C;          // 32768 elements per channel

    ChannelWiseMinMax_kernel<<<C, THREADS, 0, stream>>>(in, out, C, N);
}